// TransBlock_68831145886178
// MI455X (gfx1250) — compile-verified
//
#include <hip/hip_runtime.h>
#include <math.h>

#define NN 10000      // nodes
#define NE 100000     // edges
#define NH 8          // heads
#define CH 128        // channels/head
#define HC 1024       // NH*CH
#define PE 128        // node feat dim
#define PD 64         // edge feat dim
#define NB 32         // batches
#define HPD 512       // NH*PD
#define NEG_SLOPE 0.01f
#define THREADS 256

typedef __attribute__((ext_vector_type(2))) float v2f;
typedef __attribute__((ext_vector_type(8))) float v8f;

// ---------------------------------------------------------------- utilities
__global__ void fill_kernel(float* __restrict__ p, long n, float v) {
    long i = (long)blockIdx.x * blockDim.x + threadIdx.x;
    if (i < n) p[i] = v;
}

__device__ __forceinline__ void atomicMaxF(float* addr, float val) {
    int* ai = (int*)addr;
    int old = *ai;
    while (__int_as_float(old) < val) {
        int assumed = old;
        old = atomicCAS(ai, assumed, __float_as_int(val));
        if (old == assumed) break;
    }
}

// ------------------------------------------------- generic fp32 WMMA GEMM
// C[z] = A[z] @ B[z] (+bias) (+C when accum). Each wave computes a 16x64
// strip (4 N-tiles): one A fragment feeds 4 V_WMMA_F32_16X16X4_F32 per
// K-step. BTRANS is a template param so the inner loop is branch-free.
// Requires: M%16==0, (Nn/16)%4==0, K%4==0 (all true for this workload).
template <int BTRANS>
__global__ void wmma_gemm_f32(const float* __restrict__ A, long aZ, int lda,
                              const float* __restrict__ B, long bZ, int ldb,
                              const float* __restrict__ bias,
                              float* __restrict__ Cm, long cZ, int ldc,
                              int M, int Nn, int K, int accum)
{
    constexpr int TN = 4;   // N-tiles per wave

    A  += (long)blockIdx.z * aZ;
    B  += (long)blockIdx.z * bZ;
    Cm += (long)blockIdx.z * cZ;

    const int wave = threadIdx.x >> 5;
    const int lane = threadIdx.x & 31;
    const int half = lane >> 4;     // 0: lanes 0-15, 1: lanes 16-31
    const int r    = lane & 15;

    const int tilesN  = Nn >> 4;
    const int groupsN = tilesN / TN;
    const int totalGroups = (M >> 4) * groupsN;
    const int grp = blockIdx.x * (blockDim.x >> 5) + wave;
    if (grp >= totalGroups) return;   // wave-uniform: EXEC stays all-ones

    const int tm  = grp / groupsN;
    const int gn  = grp - tm * groupsN;
    const int arow  = (tm << 4) + r;
    const int bcol0 = gn * (TN * 16) + r;

    v8f acc[TN];
    if (accum) {
#pragma unroll
        for (int t = 0; t < TN; ++t)
#pragma unroll
            for (int j = 0; j < 8; ++j)
                acc[t][j] = Cm[(long)((tm << 4) + j + (half << 3)) * ldc + bcol0 + t * 16];
    } else {
#pragma unroll
        for (int t = 0; t < TN; ++t)
#pragma unroll
            for (int j = 0; j < 8; ++j) acc[t][j] = 0.0f;
    }

    const float* Arow = A + (long)arow * lda + 2 * half;
    for (int k = 0; k < K; k += 4) {
        // A 16x4 layout: VGPR i holds K = i + 2*half, M = lane%16 (contig b64)
        v2f a = *(const v2f*)(Arow + k);
#pragma unroll
        for (int t = 0; t < TN; ++t) {
            v2f b;
            const int bcol = bcol0 + t * 16;
            if (BTRANS) {
                b = *(const v2f*)(B + (long)bcol * ldb + k + 2 * half);
            } else {
                b.x = B[(long)(k + 2 * half) * ldb + bcol];
                b.y = B[(long)(k + 2 * half + 1) * ldb + bcol];
            }
            acc[t] = __builtin_amdgcn_wmma_f32_16x16x4_f32(
                false, a, false, b, (short)0, acc[t], false, false);
        }
    }

#pragma unroll
    for (int t = 0; t < TN; ++t) {
        const int bcol = bcol0 + t * 16;
        const float bv = bias ? bias[bcol] : 0.0f;
#pragma unroll
        for (int j = 0; j < 8; ++j)
            Cm[(long)((tm << 4) + j + (half << 3)) * ldc + bcol] = acc[t][j] + bv;
    }
}

// ------------------------------------------------------------ edge kernels
// alpha[e,h] = (q[dst]·k[src] + attr[e]·G[dst]) / sqrt(C); segment max via CAS
__global__ void edge_alpha_kernel(const float* __restrict__ q, const float* __restrict__ k,
                                  const float* __restrict__ G, const float* __restrict__ attr,
                                  const int* __restrict__ src, const int* __restrict__ dst,
                                  float* __restrict__ alpha, float* __restrict__ amax)
{
    int idx = blockIdx.x * blockDim.x + threadIdx.x;
    if (idx >= NE * NH) return;
    int e = idx >> 3, h = idx & 7;
    int s = src[e], d = dst[e];
    const float4* qd = (const float4*)(q + (long)d * HC + h * CH);
    const float4* ks = (const float4*)(k + (long)s * HC + h * CH);
    float acc = 0.f;
#pragma unroll 8
    for (int i = 0; i < CH / 4; ++i) {
        float4 a = qd[i], b = ks[i];
        acc += a.x * b.x + a.y * b.y + a.z * b.z + a.w * b.w;
    }
    const float4* ar = (const float4*)(attr + (long)e * PD);
    const float4* Gd = (const float4*)(G + (long)d * HPD + h * PD);
#pragma unroll 4
    for (int i = 0; i < PD / 4; ++i) {
        float4 a = ar[i], b = Gd[i];
        acc += a.x * b.x + a.y * b.y + a.z * b.z + a.w * b.w;
    }
    float al = acc * 0.08838834764831845f;  // 1/sqrt(128)
    alpha[idx] = al;
    atomicMaxF(&amax[d * NH + h], al);
}

// w_un = exp(alpha - amax[dst]); in-place; denom[dst] += w_un
__global__ void edge_exp_kernel(const int* __restrict__ dst, float* __restrict__ alpha,
                                const float* __restrict__ amax, float* __restrict__ denom)
{
    int idx = blockIdx.x * blockDim.x + threadIdx.x;
    if (idx >= NE * NH) return;
    int e = idx >> 3, h = idx & 7;
    int d = dst[e];
    float w = expf(alpha[idx] - amax[d * NH + h]);
    alpha[idx] = w;
    atomicAdd(&denom[d * NH + h], w);
}

// acc[dst,h,:] += w * v[src,h,:]   (thread = (e,h,c-quad))
__global__ void scatter_v_kernel(const int* __restrict__ src, const int* __restrict__ dst,
                                 const float* __restrict__ alpha, const float* __restrict__ denom,
                                 const float* __restrict__ v, float* __restrict__ acc)
{
    long idx = (long)blockIdx.x * blockDim.x + threadIdx.x;
    if (idx >= (long)NE * NH * (CH / 4)) return;
    int g  = (int)(idx & 31);            // CH/4 = 32
    long eh = idx >> 5;
    int e = (int)(eh >> 3), h = (int)(eh & 7);
    int s = src[e], d = dst[e];
    float w = alpha[eh] / (denom[d * NH + h] + 1e-16f);
    float4 vv = ((const float4*)(v + (long)s * HC + h * CH))[g];
    float* ap = acc + (long)d * HC + h * CH + g * 4;
    atomicAdd(ap + 0, w * vv.x);
    atomicAdd(ap + 1, w * vv.y);
    atomicAdd(ap + 2, w * vv.z);
    atomicAdd(ap + 3, w * vv.w);
}

// Aacc[dst,h,:] += w * attr[e,:]   (thread = (e,h,d-quad))
__global__ void scatter_attr_kernel(const int* __restrict__ dst,
                                    const float* __restrict__ alpha, const float* __restrict__ denom,
                                    const float* __restrict__ attr, float* __restrict__ Aacc)
{
    long idx = (long)blockIdx.x * blockDim.x + threadIdx.x;
    if (idx >= (long)NE * NH * (PD / 4)) return;
    int g  = (int)(idx & 15);            // PD/4 = 16
    long eh = idx >> 4;
    int e = (int)(eh >> 3), h = (int)(eh & 7);
    int d = dst[e];
    float w = alpha[eh] / (denom[d * NH + h] + 1e-16f);
    float4 av = ((const float4*)(attr + (long)e * PD))[g];
    float* ap = Aacc + (long)d * HPD + h * PD + g * 4;
    atomicAdd(ap + 0, w * av.x);
    atomicAdd(ap + 1, w * av.y);
    atomicAdd(ap + 2, w * av.z);
    atomicAdd(ap + 3, w * av.w);
}

// Y0[n,c] = mean_h acc[n,h,c] + skip[n,c]
__global__ void combine_kernel(const float* __restrict__ acc, const float* __restrict__ skip,
                               float* __restrict__ Y0)
{
    int idx = blockIdx.x * blockDim.x + threadIdx.x;
    if (idx >= NN * PE) return;
    int n = idx >> 7, c = idx & 127;
    const float* a = acc + (long)n * HC + c;
    float s = 0.f;
#pragma unroll
    for (int h = 0; h < NH; ++h) s += a[h * CH];
    Y0[idx] = s * 0.125f + skip[idx];
}

// --------------------------------------------------------------- GraphNorm
__global__ void gn_pass1(const float* __restrict__ Y0, const int* __restrict__ batch,
                         float* __restrict__ mean, float* __restrict__ cnt)
{
    int idx = blockIdx.x * blockDim.x + threadIdx.x;
    if (idx >= NN * PE) return;
    int n = idx >> 7, c = idx & 127;
    int b = batch[n];
    atomicAdd(&mean[b * PE + c], Y0[idx]);
    if (c == 0) atomicAdd(&cnt[b], 1.0f);
}

__global__ void gn_pass2(const float* __restrict__ Y0, const int* __restrict__ batch,
                         const float* __restrict__ mean, const float* __restrict__ cnt,
                         const float* __restrict__ ms,
                         float* __restrict__ Yc, float* __restrict__ var)
{
    int idx = blockIdx.x * blockDim.x + threadIdx.x;
    if (idx >= NN * PE) return;
    int n = idx >> 7, c = idx & 127;
    int b = batch[n];
    float cn = fmaxf(cnt[b], 1.0f);
    float m = mean[b * PE + c] / cn;
    float o = Y0[idx] - m * ms[c];
    Yc[idx] = o;
    atomicAdd(&var[b * PE + c], o * o);
}

__global__ void gn_pass3(const float* __restrict__ Yc, const int* __restrict__ batch,
                         const float* __restrict__ var, const float* __restrict__ cnt,
                         const float* __restrict__ w, const float* __restrict__ bb,
                         float* __restrict__ out)
{
    int idx = blockIdx.x * blockDim.x + threadIdx.x;
    if (idx >= NN * PE) return;
    int n = idx >> 7, c = idx & 127;
    int b = batch[n];
    float cn = fmaxf(cnt[b], 1.0f);
    float vv = var[b * PE + c] / cn;
    float y = Yc[idx] * (1.0f / sqrtf(vv + 1e-5f)) * w[c] + bb[c];
    out[idx] = (y > 0.f) ? y : y * NEG_SLOPE;
}

// ------------------------------------------------------------------- host
extern "C" void kernel_launch(void* const* d_in, const int* in_sizes, int n_in,
                              void* d_out, int out_size, void* d_ws, size_t ws_size,
                              hipStream_t stream)
{
    (void)in_sizes; (void)n_in; (void)out_size; (void)ws_size;

    const float* x     = (const float*)d_in[0];
    const int*   index = (const int*)d_in[1];
    const float* attr  = (const float*)d_in[2];
    const int*   batch = (const int*)d_in[3];
    const int*   src   = index;
    const int*   dst   = index + NE;

    // weights: per layer l: 0 Wq,1 bq,2 Wk,3 bk,4 Wv,5 bv,6 We,7 Wskip,8 bskip,9 gnw,10 gnb,11 gnm
    auto W = [&](int l, int j) -> const float* { return (const float*)d_in[4 + l * 12 + j]; };

    float* ws = (float*)d_ws;
    float* q     = ws; ws += (long)NN * HC;
    float* kbuf  = ws; ws += (long)NN * HC;
    float* vbuf  = ws; ws += (long)NN * HC;
    float* acc   = ws; ws += (long)NN * HC;
    float* G     = ws; ws += (long)NN * HPD;
    float* Aacc  = ws; ws += (long)NN * HPD;
    float* skip  = ws; ws += (long)NN * PE;
    float* Y0    = ws; ws += (long)NN * PE;
    float* Yc    = ws; ws += (long)NN * PE;
    float* X2    = ws; ws += (long)NN * PE;
    float* alpha = ws; ws += (long)NE * NH;
    float* amax  = ws; ws += (long)NN * NH;
    float* denom = ws; ws += (long)NN * NH;
    float* gmean = ws; ws += NB * PE;
    float* gvar  = ws; ws += NB * PE;
    float* gcnt  = ws; ws += NB;

    auto fill = [&](float* p, long n, float v) {
        long blocks = (n + THREADS - 1) / THREADS;
        fill_kernel<<<dim3((unsigned)blocks), dim3(THREADS), 0, stream>>>(p, n, v);
    };
    auto gemm_n = [&](const float* A, long aZ, int lda,
                      const float* Bm, long bZ, int ldb,
                      const float* bias, float* Cm, long cZ, int ldc,
                      int M, int Nn, int K, int accum, int Z) {
        int groups = (M / 16) * (Nn / 16 / 4);
        int blocks = (groups + 7) / 8;   // 8 waves (one 16x64 strip each) per block
        wmma_gemm_f32<0><<<dim3(blocks, 1, Z), dim3(THREADS), 0, stream>>>(
            A, aZ, lda, Bm, bZ, ldb, bias, Cm, cZ, ldc, M, Nn, K, accum);
    };
    auto gemm_t = [&](const float* A, long aZ, int lda,
                      const float* Bm, long bZ, int ldb,
                      const float* bias, float* Cm, long cZ, int ldc,
                      int M, int Nn, int K, int accum, int Z) {
        int groups = (M / 16) * (Nn / 16 / 4);
        int blocks = (groups + 7) / 8;
        wmma_gemm_f32<1><<<dim3(blocks, 1, Z), dim3(THREADS), 0, stream>>>(
            A, aZ, lda, Bm, bZ, ldb, bias, Cm, cZ, ldc, M, Nn, K, accum);
    };

    const int EH_BLKS   = (NE * NH + THREADS - 1) / THREADS;                    // 3125
    const int NP_BLKS   = (NN * PE + THREADS - 1) / THREADS;                    // 5000
    const long SV_TOT   = (long)NE * NH * (CH / 4);
    const long SA_TOT   = (long)NE * NH * (PD / 4);
    const int SV_BLKS   = (int)((SV_TOT + THREADS - 1) / THREADS);              // 100000
    const int SA_BLKS   = (int)((SA_TOT + THREADS - 1) / THREADS);              // 50000

    for (int l = 0; l < 2; ++l) {
        const float* X = (l == 0) ? x : X2;
        float* out     = (l == 0) ? X2 : (float*)d_out;

        // init accumulators
        fill(acc,  (long)NN * HC, 0.f);
        fill(Aacc, (long)NN * HPD, 0.f);
        fill(denom,(long)NN * NH, 0.f);
        fill(amax, (long)NN * NH, -INFINITY);
        fill(gmean, NB * PE, 0.f);
        fill(gvar,  NB * PE, 0.f);
        fill(gcnt,  NB, 0.f);

        // projections (WMMA f32)
        gemm_n(X, 0, PE, W(l,0), 0, HC, W(l,1), q,    0, HC, NN, HC, PE, 0, 1);  // q
        gemm_n(X, 0, PE, W(l,2), 0, HC, W(l,3), kbuf, 0, HC, NN, HC, PE, 0, 1);  // k
        gemm_n(X, 0, PE, W(l,4), 0, HC, W(l,5), vbuf, 0, HC, NN, HC, PE, 0, 1);  // v
        gemm_n(X, 0, PE, W(l,7), 0, PE, W(l,8), skip, 0, PE, NN, PE, PE, 0, 1);  // skip
        // G[n,h,d] = q_h[n,:] @ We_h^T  (per-head, B transposed)
        gemm_t(q, 128, HC, W(l,6), 128, HC, nullptr, G, 64, HPD, NN, PD, CH, 0, NH);

        // attention
        edge_alpha_kernel<<<dim3(EH_BLKS), dim3(THREADS), 0, stream>>>(
            q, kbuf, G, attr, src, dst, alpha, amax);
        edge_exp_kernel<<<dim3(EH_BLKS), dim3(THREADS), 0, stream>>>(
            dst, alpha, amax, denom);

        // message scatter: v part + attr part
        scatter_v_kernel<<<dim3(SV_BLKS), dim3(THREADS), 0, stream>>>(
            src, dst, alpha, denom, vbuf, acc);
        scatter_attr_kernel<<<dim3(SA_BLKS), dim3(THREADS), 0, stream>>>(
            dst, alpha, denom, attr, Aacc);

        // acc[n,h,:] += Aacc[n,h,:] @ We_h   (per-head, accumulate into C)
        gemm_n(Aacc, 64, HPD, W(l,6), 128, HC, nullptr, acc, 128, HC, NN, CH, PD, 1, NH);

        // head mean + skip
        combine_kernel<<<dim3(NP_BLKS), dim3(THREADS), 0, stream>>>(acc, skip, Y0);

        // GraphNorm + LeakyReLU
        gn_pass1<<<dim3(NP_BLKS), dim3(THREADS), 0, stream>>>(Y0, batch, gmean, gcnt);
        gn_pass2<<<dim3(NP_BLKS), dim3(THREADS), 0, stream>>>(Y0, batch, gmean, gcnt,
                                                              W(l,11), Yc, gvar);
        gn_pass3<<<dim3(NP_BLKS), dim3(THREADS), 0, stream>>>(Yc, batch, gvar, gcnt,
                                                              W(l,9), W(l,10), out);
    }
}